// GenesisClassifierFast_5068061409910
// MI455X (gfx1250) — compile-verified
//
#include <hip/hip_runtime.h>
#include <hip/hip_bf16.h>
#include <math.h>

// Problem constants (match reference)
#define GAMMA_F 0.618033988749895f
#define B_SZ 4096
#define T_LEN 2048
#define H_DIM 18
#define K_EXT 20          // 18 hidden + x + bias-one  -> exactly 5 chunks of K=4
#define N_CLS 2
#define N_D 4
#define WAVES_PER_BLOCK 4
#define TPW 2             // batch tiles (16 rows each) per wave: ILP for the serial chain

typedef __attribute__((ext_vector_type(2))) float v2f;
typedef __attribute__((ext_vector_type(4))) float v4f;
typedef __attribute__((ext_vector_type(8))) float v8f;

__device__ __forceinline__ float tanh_fast(float v) {
#if __has_builtin(__builtin_amdgcn_tanhf)
    return __builtin_amdgcn_tanhf(v);     // v_tanh_f32 (gfx1250 trans op)
#else
    return tanhf(v);
#endif
}

__global__ __launch_bounds__(WAVES_PER_BLOCK * 32)
void genesis_rnn_wmma_kernel(const float* __restrict__ x,       // [B, T]
                             const float* __restrict__ w_in,    // [H]
                             const float* __restrict__ b_in,    // [H]
                             const float* __restrict__ W_hh,    // [H, H] row-major
                             const float* __restrict__ coupling,// [1]
                             const float* __restrict__ W_out,   // [H, 2]
                             const float* __restrict__ b_out,   // [2]
                             float* __restrict__ out)           // [B, 2]
{
    // Each wave owns TPW 16-row batch tiles; per-tile state = 16 rows x K_EXT cols in LDS.
    __shared__ __align__(16) float hbuf_all[WAVES_PER_BLOCK][TPW][16][K_EXT];

    const int lane = threadIdx.x & 31;
    const int wave = threadIdx.x >> 5;
    const int half = lane >> 4;      // 0: lanes 0-15, 1: lanes 16-31
    const int l16  = lane & 15;

    const int tile0 = (blockIdx.x * WAVES_PER_BLOCK + wave) * TPW;
    const float cpl = coupling[0];

    // ---- Build extended-B fragments once (held in VGPRs for all 2048 steps) ----
    // B_ext[k][n]: k<18 -> coupling*W_hh[k][n]; k==18 -> w_in[n]; k==19 -> b_in[n].
    // f32 B 4x16 fragment layout: lane holds n = nt*16 + l16, VGPR j holds k = 4c + 2*half + j.
    v2f bfrag[2][5];
    #pragma unroll
    for (int nt = 0; nt < 2; ++nt) {
        const int n = nt * 16 + l16;
        #pragma unroll
        for (int c = 0; c < 5; ++c) {
            const int k0 = 4 * c + 2 * half;
            float e0 = 0.f, e1 = 0.f;
            if (n < H_DIM) {
                e0 = (k0 < H_DIM) ? cpl * W_hh[k0 * H_DIM + n]
                   : (k0 == H_DIM ? w_in[n] : b_in[n]);
                const int k1 = k0 + 1;
                e1 = (k1 < H_DIM) ? cpl * W_hh[k1 * H_DIM + n]
                   : (k1 == H_DIM ? w_in[n] : b_in[n]);
            }
            bfrag[nt][c].x = e0;
            bfrag[nt][c].y = e1;
        }
    }

    // ---- Init state: h = 0, ones column = 1 ----
    #pragma unroll
    for (int j = 0; j < TPW; ++j)
        for (int idx = lane; idx < 16 * K_EXT; idx += 32) {
            const int r = idx / K_EXT, c = idx % K_EXT;
            hbuf_all[wave][j][r][c] = (c == K_EXT - 1) ? 1.0f : 0.0f;
        }
    __builtin_amdgcn_sched_barrier(0);   // wave-private LDS: DS ops in-order; fence the compiler only

    const float* xrow[TPW];
    #pragma unroll
    for (int j = 0; j < TPW; ++j)
        xrow[j] = x + (size_t)((tile0 + j) * 16 + l16) * T_LEN;   // lanes 0-15 use this

    // Software-pipelined x fetch: one b128 (4 steps) per tile per group, loaded a group ahead.
    v4f xcur[TPW], xnxt[TPW];
    if (half == 0) {
        #pragma unroll
        for (int j = 0; j < TPW; ++j)
            xcur[j] = *(const v4f*)&xrow[j][0];
    }

    // ---- Recurrence over T, unrolled by 4 ----
    for (int t0 = 0; t0 < T_LEN; t0 += 4) {
        if (half == 0) {
            const int tn = (t0 + 4 < T_LEN) ? (t0 + 4) : 0;   // dummy (in-range) load on last group
            #pragma unroll
            for (int j = 0; j < TPW; ++j) {
                xnxt[j] = *(const v4f*)&xrow[j][tn];
                __builtin_prefetch(&xrow[j][(t0 + 32 < T_LEN) ? (t0 + 32) : tn], 0, 3);
            }
        }

        #pragma unroll
        for (int u = 0; u < 4; ++u) {
            // x_t into extended column 18 of each tile's state
            if (half == 0) {
                #pragma unroll
                for (int j = 0; j < TPW; ++j)
                    hbuf_all[wave][j][l16][H_DIM] = xcur[j][u];
            }
            __builtin_amdgcn_sched_barrier(0);

            // A fragments: lane holds row l16, VGPR pair = k = 4c + 2*half (+1)
            v2f a[TPW][5];
            #pragma unroll
            for (int j = 0; j < TPW; ++j)
                #pragma unroll
                for (int c = 0; c < 5; ++c)
                    a[j][c] = *(const v2f*)&hbuf_all[wave][j][l16][4 * c + 2 * half];

            // pre = A_ext @ B_ext : TPW x 2 independent accumulator chains interleave
            v8f d0[TPW], d1[TPW];
            #pragma unroll
            for (int j = 0; j < TPW; ++j) { d0[j] = (v8f){}; d1[j] = (v8f){}; }
            #pragma unroll
            for (int c = 0; c < 5; ++c)
                #pragma unroll
                for (int j = 0; j < TPW; ++j) {
                    d0[j] = __builtin_amdgcn_wmma_f32_16x16x4_f32(false, a[j][c], false, bfrag[0][c],
                                                                  (short)0, d0[j], false, false);
                    d1[j] = __builtin_amdgcn_wmma_f32_16x16x4_f32(false, a[j][c], false, bfrag[1][c],
                                                                  (short)0, d1[j], false, false);
                }

            // h_new = tanh(pre); clamp units >= N_D to [-GAMMA, GAMMA].
            const bool clamp0 = (l16 >= N_D);       // tile-1 columns (16,17) always clamp
            #pragma unroll
            for (int j = 0; j < TPW; ++j)
                #pragma unroll
                for (int i = 0; i < 8; ++i) {
                    float v0  = tanh_fast(d0[j][i]);
                    float v0c = fminf(fmaxf(v0, -GAMMA_F), GAMMA_F);
                    d0[j][i]  = clamp0 ? v0c : v0;
                    float v1  = tanh_fast(d1[j][i]);
                    d1[j][i]  = fminf(fmaxf(v1, -GAMMA_F), GAMMA_F);
                }

            // Transpose D-layout -> A-layout via LDS (wave-private; DS is in-order in hardware)
            __builtin_amdgcn_sched_barrier(0);
            const int rbase = half * 8;
            #pragma unroll
            for (int j = 0; j < TPW; ++j) {
                #pragma unroll
                for (int i = 0; i < 8; ++i)
                    hbuf_all[wave][j][rbase + i][l16] = d0[j][i];
                if (l16 < H_DIM - 16) {             // columns 16,17 from the second N tile
                    #pragma unroll
                    for (int i = 0; i < 8; ++i)
                        hbuf_all[wave][j][rbase + i][16 + l16] = d1[j][i];
                }
            }
            __builtin_amdgcn_sched_barrier(0);
        }

        #pragma unroll
        for (int j = 0; j < TPW; ++j) xcur[j] = xnxt[j];
    }

    // ---- Final projection: logits = h_final @ W_out + b_out  (tiny: 18x2) ----
    if (half == 0) {
        #pragma unroll
        for (int j = 0; j < TPW; ++j) {
            float acc0 = b_out[0], acc1 = b_out[1];
            #pragma unroll
            for (int k = 0; k < H_DIM; ++k) {
                const float hk = hbuf_all[wave][j][l16][k];
                acc0 = fmaf(hk, W_out[k * N_CLS + 0], acc0);
                acc1 = fmaf(hk, W_out[k * N_CLS + 1], acc1);
            }
            float* o = out + (size_t)((tile0 + j) * 16 + l16) * N_CLS;
            o[0] = acc0;
            o[1] = acc1;
        }
    }
}

extern "C" void kernel_launch(void* const* d_in, const int* in_sizes, int n_in,
                              void* d_out, int out_size, void* d_ws, size_t ws_size,
                              hipStream_t stream) {
    (void)in_sizes; (void)n_in; (void)out_size; (void)d_ws; (void)ws_size;
    const float* x        = (const float*)d_in[0];
    const float* w_in     = (const float*)d_in[1];
    const float* b_in     = (const float*)d_in[2];
    const float* W_hh     = (const float*)d_in[3];
    const float* coupling = (const float*)d_in[4];
    const float* W_out    = (const float*)d_in[5];
    const float* b_out    = (const float*)d_in[6];
    float* out            = (float*)d_out;

    const int tiles  = B_SZ / 16;                            // 256 tiles of 16 batch rows
    const int waves  = tiles / TPW;                          // 128 waves
    const int blocks = waves / WAVES_PER_BLOCK;              // 32 blocks x 128 threads
    genesis_rnn_wmma_kernel<<<blocks, WAVES_PER_BLOCK * 32, 0, stream>>>(
        x, w_in, b_in, W_hh, coupling, W_out, b_out, out);
}